// GCNModel_12412455485983
// MI455X (gfx1250) — compile-verified
//
#include <hip/hip_runtime.h>
#include <hip/hip_bf16.h>

// ---------------------------------------------------------------------------
// 3-layer GCN (PyG GCNConv semantics) for MI455X / gfx1250.
//
// Roofline: dense GEMMs are only ~8.2 GFLOP; the edge gather + atomic scatter
// (~1.6 GB of L2 traffic per 128-wide layer) dominates. T (51.2MB) and agg
// (51.2MB) both fit in the 192MB L2, so scatter runs at L2-atomic speed.
// Since the kernel is L2-bound, we keep full fp32 precision and use
// V_WMMA_F32_16X16X4_F32 for the feature transforms.
//
// Round-2 refinement: W is staged into LDS pre-swizzled into the B-operand
// layout (even/odd K rows interleaved per column), so each WMMA B operand is
// one aligned ds_load_b64 — removing the v_dual_mov repacking seen in the
// round-1 assembly. Pair-row pitch is padded so lanes 0-15 / 16-31 land on
// disjoint LDS bank halves.
// ---------------------------------------------------------------------------

#define CIN 128

typedef __attribute__((ext_vector_type(2))) float v2f;
typedef __attribute__((ext_vector_type(4))) float v4f;
typedef __attribute__((ext_vector_type(8))) float v8f;

// ---------------- degree / normalization ----------------

__global__ void fill_deg(float* __restrict__ deg, int N) {
  int i = blockIdx.x * blockDim.x + threadIdx.x;
  if (i < N) deg[i] = 1.0f;  // self-loop weight contributes 1 to deg
}

__global__ void deg_edges(const int* __restrict__ dst, const float* __restrict__ ea,
                          float* __restrict__ deg, int E) {
  int e = blockIdx.x * blockDim.x + threadIdx.x;
  if (e < E) atomicAdd(&deg[dst[e]], ea[e]);
}

__global__ void calc_dinv(const float* __restrict__ deg, float* __restrict__ dinv, int N) {
  int i = blockIdx.x * blockDim.x + threadIdx.x;
  if (i < N) {
    float d = deg[i];
    dinv[i] = (d > 0.0f) ? rsqrtf(d) : 0.0f;
  }
}

// ---------------- fp32 WMMA GEMM: T[N,COUT] = act(H[N,128]) @ W[128,COUT] ----
//
// One wave per 16-row strip; wave holds COUT/16 accumulator tiles (v8f each)
// and iterates K in steps of 4 using V_WMMA_F32_16X16X4_F32.
//
// fp32 A layout (16x4): lane = m + 16*(k>>1), vgpr = k&1  -> per-lane float2.
// fp32 B layout (4x16): mirrored: lane = n + 16*(k>>1), vgpr = k&1.
// fp32 C/D layout (16x16): vgpr v: lanes 0-15 -> row v, lanes 16-31 -> row v+8.
//
// LDS staging layout for W ("B-operand layout"):
//   pair p = k/2 holds K rows {2p, 2p+1}; element (k,n) lives at
//   sW[p*PITCH + 2*n + (k&1)]  ->  lane's B operand = one aligned 8B load.
//   PITCH = 2*COUT + 32 words  =>  PITCH % 64 == 32, so the kh=1 half-wave
//   reads the opposite bank half from kh=0: conflict-free b64 LDS reads.
//
// RELU fuses the previous layer's activation into the A load (bias was folded
// into agg by the scatter-init kernel).

template <int COUT, bool RELU>
__global__ __launch_bounds__(256) void gemm_wmma(const float* __restrict__ H,
                                                 const float* __restrict__ Wt,
                                                 float* __restrict__ T,
                                                 int nRowTiles) {
  constexpr int PITCH = 2 * COUT + 32;           // words per K-row pair
  __shared__ float sW[(CIN / 2) * PITCH];        // 72KB (COUT=128) / 40KB (64)

  // Cooperative staging of W into the interleaved B-operand layout.
  for (int i = threadIdx.x; i < CIN * COUT; i += blockDim.x) {
    int k = i / COUT;
    int n = i - k * COUT;
    sW[(k >> 1) * PITCH + 2 * n + (k & 1)] = Wt[i];
  }
  __syncthreads();

  const int wave = threadIdx.x >> 5;
  const int lane = threadIdx.x & 31;
  const int rt   = blockIdx.x * 8 + wave;
  if (rt >= nRowTiles) return;  // wave-uniform: EXEC stays all-ones for WMMA

  const int m0 = rt * 16;
  const int m  = lane & 15;    // row within tile (A) / col within tile (B,D)
  const int kh = lane >> 4;    // K half-select

  constexpr int NT = COUT / 16;
  v8f acc[NT] = {};

  const float* arow = H + (size_t)(m0 + m) * CIN + 2 * kh;

  for (int k = 0; k < CIN; k += 4) {
    v2f a = *(const v2f*)(arow + k);  // H[m0+m, k+2*kh .. +1], 8B aligned
    if (RELU) {
      a.x = fmaxf(a.x, 0.0f);
      a.y = fmaxf(a.y, 0.0f);
    }
    // B pair row index: (k + 2*kh)/2 = k/2 + kh
    const float* b0 = &sW[((k >> 1) + kh) * PITCH + 2 * m];
#pragma unroll
    for (int t = 0; t < NT; ++t) {
      v2f b = *(const v2f*)(b0 + 32 * t);  // {W[k+2kh, n], W[k+2kh+1, n]}
      acc[t] = __builtin_amdgcn_wmma_f32_16x16x4_f32(
          /*neg_a=*/false, a, /*neg_b=*/false, b,
          /*c_mod=*/(short)0, acc[t], /*reuse_a=*/false, /*reuse_b=*/false);
    }
  }

  // D layout: vgpr v -> row m0 + v + 8*kh, col t*16 + m
  float* out = T + (size_t)(m0 + 8 * kh) * COUT + m;
#pragma unroll
  for (int t = 0; t < NT; ++t) {
#pragma unroll
    for (int v = 0; v < 8; ++v) {
      out[(size_t)v * COUT + t * 16] = acc[t][v];
    }
  }
}

// ---------------- scatter-init: agg[n] = b + dinv[n]^2 * T[n] (self loop) ----

__global__ void init_agg(const float* __restrict__ T, const float* __restrict__ dinv,
                         const float* __restrict__ bias, float* __restrict__ agg,
                         int N, int cpr /* = C/4 */) {
  int i = blockIdx.x * blockDim.x + threadIdx.x;
  int total = N * cpr;
  if (i >= total) return;
  int n  = i / cpr;
  int c4 = i - n * cpr;
  float s = dinv[n];
  s = s * s;  // self-loop norm = dinv[n] * 1 * dinv[n]
  v4f t = ((const v4f*)T)[(size_t)n * cpr + c4];
  v4f b = ((const v4f*)bias)[c4];
  ((v4f*)agg)[(size_t)n * cpr + c4] = b + s * t;
}

// ---------------- edge scatter: agg[dst] += norm * T[src] -------------------
// C/4 lanes per edge; each lane gathers one float4 of the source row (L2 hit,
// T is L2-resident) and issues 4 global_atomic_add_f32 into the dst row.

template <int C>
__global__ void edge_scatter(const int* __restrict__ src, const int* __restrict__ dst,
                             const float* __restrict__ ea, const float* __restrict__ dinv,
                             const float* __restrict__ T, float* __restrict__ agg, int E) {
  constexpr int LPE = C / 4;  // 32 (C=128) or 16 (C=64) lanes per edge
  int tid  = blockIdx.x * blockDim.x + threadIdx.x;
  int e    = tid / LPE;
  int lane = tid % LPE;
  if (e >= E) return;

  int s = src[e];
  int d = dst[e];
  float nrm = dinv[s] * ea[e] * dinv[d];

  v4f v = ((const v4f*)(T + (size_t)s * C))[lane];
  float* a = agg + (size_t)d * C + lane * 4;
  atomicAdd(a + 0, nrm * v.x);
  atomicAdd(a + 1, nrm * v.y);
  atomicAdd(a + 2, nrm * v.z);
  atomicAdd(a + 3, nrm * v.w);
}

// ---------------- launch ----------------------------------------------------

extern "C" void kernel_launch(void* const* d_in, const int* in_sizes, int n_in,
                              void* d_out, int out_size, void* d_ws, size_t ws_size,
                              hipStream_t stream) {
  const float* x  = (const float*)d_in[0];
  const int*   ei = (const int*)d_in[1];
  const float* ea = (const float*)d_in[2];
  const float* W1 = (const float*)d_in[3];
  const float* b1 = (const float*)d_in[4];
  const float* W2 = (const float*)d_in[5];
  const float* b2 = (const float*)d_in[6];
  const float* W3 = (const float*)d_in[7];
  const float* b3 = (const float*)d_in[8];

  const int N = in_sizes[0] / CIN;  // 100000
  const int E = in_sizes[2];        // 1600000
  const int* srcIdx = ei;           // edge_index[0]
  const int* dstIdx = ei + E;       // edge_index[1]

  // Workspace layout: T (N*128 f32) | agg (N*128 f32) | deg (N) | dinv (N)
  float* T    = (float*)d_ws;
  float* agg  = T + (size_t)N * 128;
  float* deg  = agg + (size_t)N * 128;
  float* dinv = deg + N;
  float* out  = (float*)d_out;

  const int nRowTiles = N / 16;  // 6250 (N is a multiple of 16)
  dim3 blk(256);
  dim3 gN((N + 255) / 256);
  dim3 gE((E + 255) / 256);
  dim3 gGemm((nRowTiles + 7) / 8);
  dim3 gInit128(((unsigned)N * 32 + 255) / 256);
  dim3 gInit64(((unsigned)N * 16 + 255) / 256);
  dim3 gScat128(((unsigned)E * 32 + 255) / 256);
  dim3 gScat64(((unsigned)E * 16 + 255) / 256);

  // --- degrees + symmetric normalization ---
  fill_deg<<<gN, blk, 0, stream>>>(deg, N);
  deg_edges<<<gE, blk, 0, stream>>>(dstIdx, ea, deg, E);
  calc_dinv<<<gN, blk, 0, stream>>>(deg, dinv, N);

  // --- layer 1: h1 = relu(conv(x, W1) + b1)  (relu deferred to next GEMM) ---
  gemm_wmma<128, false><<<gGemm, blk, 0, stream>>>(x, W1, T, nRowTiles);
  init_agg<<<gInit128, blk, 0, stream>>>(T, dinv, b1, agg, N, 32);
  edge_scatter<128><<<gScat128, blk, 0, stream>>>(srcIdx, dstIdx, ea, dinv, T, agg, E);

  // --- layer 2 ---
  gemm_wmma<128, true><<<gGemm, blk, 0, stream>>>(agg, W2, T, nRowTiles);
  init_agg<<<gInit128, blk, 0, stream>>>(T, dinv, b2, agg, N, 32);
  edge_scatter<128><<<gScat128, blk, 0, stream>>>(srcIdx, dstIdx, ea, dinv, T, agg, E);

  // --- layer 3: logits straight into d_out (C=64) ---
  gemm_wmma<64, true><<<gGemm, blk, 0, stream>>>(agg, W3, T, nRowTiles);
  init_agg<<<gInit64, blk, 0, stream>>>(T, dinv, b3, out, N, 16);
  edge_scatter<64><<<gScat64, blk, 0, stream>>>(srcIdx, dstIdx, ea, dinv, T, out, E);
}